// GPT_26560077758513
// MI455X (gfx1250) — compile-verified
//
#include <hip/hip_runtime.h>
#include <math.h>

// -----------------------------------------------------------------------------
// Log-space stack machine, one thread per (b, t) element.
//   - stack (32 x sign, 32 x log) fully register resident (loop fully unrolled,
//     all stack indices are compile-time constants)
//   - incremental sum-of-squares for the RMS rescale (O(1) per step)
//   - op-probability stream (5 f32 / thread / step, dominant memory traffic)
//     double-buffered through LDS with gfx1250 async-to-LDS DMA:
//         global_load_async_to_lds_b32  (+ s_wait_asynccnt pipelining)
//     Per-thread LDS stride = 5 dwords -> bank-conflict-free (gcd(5,64)=1).
//   - native gfx1250 TRANS ops: v_tanh_f32 (inline asm, with the ISA-required
//     post-TRANS v_nop), v_exp_f32 (__expf), v_log_f32 (__logf), v_rsq_f32.
//     ~9 TRANS + ~120 VALU per step instead of ~240 VALU with ocml tanh/log1p.
// -----------------------------------------------------------------------------

#define LOG_LIM 10.0f
constexpr int NSTACK = 32;
constexpr int DEPTH  = 31;   // NSTACK - 1
constexpr int BLOCK  = 256;

// Native gfx1250 tanh. The compiler cannot model hazards of asm TRANS ops, so
// the ISA rule "1 independent op or V_NOP after a TRANS op before its output
// is used" is satisfied inside the blob. Pure (non-volatile) asm so identical
// calls can be CSE'd and the blob can be scheduled freely.
__device__ __forceinline__ float fast_tanh(float x) {
    float r;
    asm("v_tanh_f32 %0, %1\n\tv_nop" : "=v"(r) : "v"(x));
    return r;
}

__device__ __forceinline__ float fast_rsq(float x) {
#if __has_builtin(__builtin_amdgcn_rsqf)
    return __builtin_amdgcn_rsqf(x);           // v_rsq_f32
#else
    return rsqrtf(x);
#endif
}

__device__ __forceinline__ float clipLog(float x) {
    return LOG_LIM * fast_tanh(x * (1.0f / LOG_LIM));
}
__device__ __forceinline__ float sgnf(float x) {
    return (x > 0.0f) ? 1.0f : ((x < 0.0f) ? -1.0f : 0.0f);
}

// Stage 5 consecutive dwords (one step's op probs for this lane) into LDS
// asynchronously. Tracked by ASYNCcnt; async loads complete in order. The
// instruction immediate offset applies to BOTH the global and LDS addresses,
// so one (lds, global) base pair stages 5 consecutive dwords.
__device__ __forceinline__ void stage_async5(unsigned lds_off, const float* g) {
    unsigned long long ga = (unsigned long long)(size_t)g;
    asm volatile(
        "global_load_async_to_lds_b32 %0, %1, off\n\t"
        "global_load_async_to_lds_b32 %0, %1, off offset:4\n\t"
        "global_load_async_to_lds_b32 %0, %1, off offset:8\n\t"
        "global_load_async_to_lds_b32 %0, %1, off offset:12\n\t"
        "global_load_async_to_lds_b32 %0, %1, off offset:16"
        :: "v"(lds_off), "v"(ga) : "memory");
}

__global__ __launch_bounds__(BLOCK) void GPT_logstack_kernel(
    const float* __restrict__ g_sgn,   // (B,T,32)
    const float* __restrict__ g_log,   // (B,T,32)
    const float* __restrict__ g_ops,   // (B,T,31,5) softmaxed
    float* __restrict__ g_out,         // (2,B,T)
    int n)                             // n = B*T
{
    __shared__ float opsbuf[2][BLOCK * 5];   // 10240 B, double buffer

    const int tid = threadIdx.x;
    const int idx = blockIdx.x * BLOCK + tid;
    if (idx >= n) return;

    // AS3->generic addrspacecast keeps the LDS byte offset in the low 32 bits,
    // which is exactly what the async-to-LDS VDST operand wants.
    const unsigned loff0 = (unsigned)(size_t)(&opsbuf[0][tid * 5]);
    const unsigned loff1 = (unsigned)(size_t)(&opsbuf[1][tid * 5]);

    const float* gops = g_ops + (size_t)idx * (DEPTH * 5);

    // ---- prologue: stage step 0 into buffer 0 as early as possible, so the
    // async DMA overlaps the stack loads below.
    stage_async5(loff0, gops);

    // ---- load the 32-deep (sign, log) stack: 8x b128 per array, lane stride
    // 128 B -> each wave transaction covers 4 KB of contiguous memory.
    float s[NSTACK], l[NSTACK];
    {
        const float4* sg4 = reinterpret_cast<const float4*>(g_sgn + (size_t)idx * NSTACK);
        const float4* lg4 = reinterpret_cast<const float4*>(g_log + (size_t)idx * NSTACK);
#pragma unroll
        for (int i = 0; i < NSTACK / 4; ++i) {
            float4 a = sg4[i];
            float4 b = lg4[i];
            s[4 * i + 0] = a.x; s[4 * i + 1] = a.y; s[4 * i + 2] = a.z; s[4 * i + 3] = a.w;
            l[4 * i + 0] = b.x; l[4 * i + 1] = b.y; l[4 * i + 2] = b.z; l[4 * i + 3] = b.w;
        }
    }

    // running sum of squares of the current stack's log values
    float S = 0.0f;
#pragma unroll
    for (int i = 0; i < NSTACK; ++i) S += l[i] * l[i];

#pragma unroll
    for (int step = 0; step < DEPTH; ++step) {
        const int cs  = NSTACK - step;   // current stack size (compile-time)
        const int top = cs - 1;
        const int sec = cs - 2;
        const int buf = step & 1;

        // stage next step's probs while we still owe a wait for this step's
        if (step + 1 < DEPTH) {
            const unsigned lnext = ((step + 1) & 1) ? loff1 : loff0;
            // make sure the LDS reads of the buffer we are about to overwrite
            // (two steps ago) have fully retired before the DMA can land
            asm volatile("s_wait_dscnt 0" ::: "memory");
            stage_async5(lnext, gops + (step + 1) * 5);
            // 10 outstanding async b32s; they complete in order, so <=5
            // guarantees this step's 5 have landed in LDS.
            asm volatile("s_wait_asynccnt 5" ::: "memory");
        } else {
            asm volatile("s_wait_asynccnt 0" ::: "memory");
        }

        const float* pb = &opsbuf[buf][tid * 5];
        const float p0 = pb[0], p1 = pb[1], p2 = pb[2], p3 = pb[3], p4 = pb[4];

        const float s1 = s[sec], L1 = l[sec];   // second-from-top (x)
        const float s2 = s[top], L2 = l[top];   // top (y)

        // ---- shared pieces between add/sub
        const float amax = fmaxf(L1, L2);
        const float amin = fminf(L1, L2);
        const float e1   = __expf(amin - amax);          // in (0, 1]
        const float la     = amax + __logf(1.0f + e1);   // logaddexp
        const float l_same = clipLog(la);

        const bool  bx    = (L1 >= L2);
        const float delta = fminf(fmaxf(amin - amax, -LOG_LIM), -0.001f);
        const float e2    = __expf(delta);               // in [4.5e-5, 0.999]
        const float diff  = __logf(1.0f - e2);           // log1p(-e2)
        const bool  zres  = (amin == amax);
        const float l_opp = zres ? 0.0f : (amax + diff);

        const bool  zx = (s1 == 0.0f);
        const bool  zy = (s2 == 0.0f);
        const bool  only_x = (!zx) && zy;
        const bool  only_y = zx && (!zy);
        const bool  both   = (!zx) && (!zy);
        const float p12    = s1 * s2;

        // ---- add_log_space(x, y)
        float addS, addL;
        {
            const bool  same = both && (p12 > 0.0f);
            const bool  opp  = both && !(p12 > 0.0f);
            const float bsgn = bx ? s1 : s2;
            float so = only_x ? s1 : 0.0f;
            float lo = only_x ? L1 : 0.0f;
            so = only_y ? s2 : so;  lo = only_y ? L2 : lo;
            so = same ? sgnf(s1) : so;  lo = same ? l_same : lo;
            so = opp ? (zres ? 0.0f : bsgn) : so;  lo = opp ? l_opp : lo;
            addS = so;
            addL = clipLog(lo);
        }
        // ---- subtract = add_log_space(x, (-sy, ly))
        float subS, subL;
        {
            const bool  same = both && (p12 < 0.0f);   // s1*(-s2) > 0
            const bool  opp  = both && !(p12 < 0.0f);
            const float bsgn = bx ? s1 : -s2;
            float so = only_x ? s1 : 0.0f;
            float lo = only_x ? L1 : 0.0f;
            so = only_y ? -s2 : so;  lo = only_y ? L2 : lo;
            so = same ? sgnf(s1) : so;  lo = same ? l_same : lo;
            so = opp ? (zres ? 0.0f : bsgn) : so;  lo = opp ? l_opp : lo;
            subS = so;
            subL = clipLog(lo);
        }
        const float mulS = p12, mulL = clipLog(L1 + L2);
        const float divS = p12, divL = clipLog(L1 - L2);

        // soft selection over [add, sub, mul, div, identity]
        float resS = p0 * addS + p1 * subS + p2 * mulS + p3 * divS + p4 * s1;
        float resL = p0 * addL + p1 * subL + p2 * mulL + p3 * divL + p4 * L1;

        // ---- rms rescale over [old stack below top, resL] (cs entries)
        // scale = min(LOG_LIM / sqrt(m), 1) = min(LOG_LIM * rsqrt(m), 1)
        const float Sprev = S - L2 * L2;   // sum over indices 0..cs-2 (old values)
        const float m     = (Sprev + resL * resL) * (1.0f / (float)cs) + 1e-6f;
        const float scale = fminf(LOG_LIM * fast_rsq(m), 1.0f);
        resL *= scale;

        // update running sum-of-squares for the new (shrunk) stack
        S = Sprev - L1 * L1 + resL * resL;
        s[sec] = resS;
        l[sec] = resL;
    }

    // output: plane 0 = final sign, plane 1 = final log
    g_out[idx]             = s[0];
    g_out[(size_t)n + idx] = l[0];
}

extern "C" void kernel_launch(void* const* d_in, const int* in_sizes, int n_in,
                              void* d_out, int out_size, void* d_ws, size_t ws_size,
                              hipStream_t stream) {
    const float* g_sgn = (const float*)d_in[0];   // (B,T,32) f32
    const float* g_log = (const float*)d_in[1];   // (B,T,32) f32
    const float* g_ops = (const float*)d_in[2];   // (B,T,31,5) f32
    float* out = (float*)d_out;                   // (2,B,T) f32

    const int n    = in_sizes[0] / NSTACK;        // B*T
    const int grid = (n + BLOCK - 1) / BLOCK;

    GPT_logstack_kernel<<<grid, BLOCK, 0, stream>>>(g_sgn, g_log, g_ops, out, n);
}